// QuantumQLSTM_65481071406634
// MI455X (gfx1250) — compile-verified
//
#include <hip/hip_runtime.h>
#include <hip/hip_bf16.h>
#include <math.h>

// ---------------------------------------------------------------------------
// QLSTM collapse: gates are scalars => hx/cx uniform across D_H.
//   Z[t,0:32] = X[t,:] @ Wx.T + b        (parallel GEMM, WMMA f32 16x16x4)
//   scalar recurrence over t (1 wave)
//   broadcast writeout of stacked/hx/cx  (b128 stores)
// ---------------------------------------------------------------------------

typedef float v2f __attribute__((ext_vector_type(2)));
typedef float v8f __attribute__((ext_vector_type(8)));

#define T_STEPS 8192
#define D_IN    2048
#define D_H     2048
#define D_TOT   4096
#define NG      32          // 4 gates * 8 wires

// workspace layout (float offsets)
#define WS_Z   0                         // [T_STEPS][NG]         1 MB
#define WS_WP  (WS_Z + T_STEPS * NG)     // [NG][D_IN] packed Wx  256 KB
#define WS_S   (WS_WP + NG * D_IN)       // [NG] rowsum of Wh part
#define WS_B   (WS_S + NG)               // [NG] biases
#define WS_H   (WS_B + NG)               // [T_STEPS] h_t scalars
#define WS_C   (WS_H + T_STEPS)          // final c scalar

// ---------------------------------------------------------------------------
// Prep: pack W[:, :2048] of all 4 gates into Wp[32][2048] (gate-major),
// compute s[r] = sum_k W[r][2048+k], stash biases.
// ---------------------------------------------------------------------------
__global__ __launch_bounds__(256)
void qlstm_prep(const float* __restrict__ Wf, const float* __restrict__ bf,
                const float* __restrict__ Wi, const float* __restrict__ bi,
                const float* __restrict__ Wu, const float* __restrict__ bu,
                const float* __restrict__ Wo, const float* __restrict__ bo,
                float* __restrict__ ws) {
    int r    = blockIdx.x;        // 0..31
    int gate = r >> 3;
    int row  = r & 7;
    const float* W = (gate == 0) ? Wf : (gate == 1) ? Wi : (gate == 2) ? Wu : Wo;
    const float* b = (gate == 0) ? bf : (gate == 1) ? bi : (gate == 2) ? bu : bo;
    const float* src = W + (size_t)row * D_TOT;
    float* wp = ws + WS_WP + (size_t)r * D_IN;

    float partial = 0.f;
    for (int k = threadIdx.x; k < D_IN; k += blockDim.x) {
        wp[k] = src[k];                 // x-part of the weight row
        partial += src[D_IN + k];       // rowsum of h-part (h is uniform)
    }
    __shared__ float red[256];
    red[threadIdx.x] = partial;
    __syncthreads();
    for (int off = 128; off > 0; off >>= 1) {
        if (threadIdx.x < off) red[threadIdx.x] += red[threadIdx.x + off];
        __syncthreads();
    }
    if (threadIdx.x == 0) {
        ws[WS_S + r] = red[0];
        ws[WS_B + r] = b[row];
    }
}

// ---------------------------------------------------------------------------
// GEMM: Z = X @ Wp.T + bias.  M=8192, K=2048, N=32.
// Block = 128 threads (4 waves). Each block owns a 16x32 output tile;
// each wave reduces a K-chunk of 512 with V_WMMA_F32_16X16X4_F32, partials
// summed through LDS.
// A (16x4 f32): lane l -> row l&15, VGPR v -> K = v + 2*(l>>4)
// B (4x16 f32): lane l -> col l&15, VGPR v -> K = v + 2*(l>>4)   (B[k][n]=Wp[n][k])
// C (16x16 f32): lane l, VGPR v -> M = v + 8*(l>>4), N = l&15
// ---------------------------------------------------------------------------
__global__ __launch_bounds__(128)
void qlstm_gemm(const float* __restrict__ X, float* __restrict__ ws) {
    const float* Wp   = ws + WS_WP;
    const float* bias = ws + WS_B;
    float* Z = ws + WS_Z;

    const int t0    = blockIdx.x * 16;
    const int tid   = threadIdx.x;
    const int wave  = tid >> 5;
    const int lane  = tid & 31;
    const int r     = lane & 15;
    const int khalf = (lane >> 4) * 2;          // 0 or 2
    const int kbase = wave * (D_IN / 4);        // 512 per wave

    const float* xrow  = X  + (size_t)(t0 + r) * D_IN;
    const float* w0row = Wp + (size_t)r        * D_IN;
    const float* w1row = Wp + (size_t)(16 + r) * D_IN;

    v8f acc0 = {0.f,0.f,0.f,0.f,0.f,0.f,0.f,0.f};
    v8f acc1 = {0.f,0.f,0.f,0.f,0.f,0.f,0.f,0.f};

    #pragma unroll 4
    for (int k = kbase; k < kbase + D_IN / 4; k += 4) {
        const int ko = k + khalf;
        v2f a  = *(const v2f*)(xrow  + ko);
        v2f b0 = *(const v2f*)(w0row + ko);
        v2f b1 = *(const v2f*)(w1row + ko);
        acc0 = __builtin_amdgcn_wmma_f32_16x16x4_f32(
            false, a, false, b0, (short)0, acc0, false, false);
        acc1 = __builtin_amdgcn_wmma_f32_16x16x4_f32(
            false, a, false, b1, (short)0, acc1, false, false);
    }

    __shared__ float tile[4][16][32];
    const int mb = (lane >> 4) * 8;
    #pragma unroll
    for (int v = 0; v < 8; ++v) {
        tile[wave][mb + v][r]      = acc0[v];
        tile[wave][mb + v][16 + r] = acc1[v];
    }
    __syncthreads();

    #pragma unroll
    for (int e = tid; e < 512; e += 128) {
        int m = e >> 5, n = e & 31;
        float sum = tile[0][m][n] + tile[1][m][n] +
                    tile[2][m][n] + tile[3][m][n] + bias[n];
        Z[(size_t)(t0 + m) * NG + n] = sum;
    }
}

// ---------------------------------------------------------------------------
// Scalar recurrence, one wave32. Lane l owns angle column l (gate = l/8).
// qexp: product of cos^2(ang/2) within each 8-lane gate group (shfl_xor),
// then cross-gate broadcast. All lanes keep (h,c) replicated.
// ---------------------------------------------------------------------------
__global__ __launch_bounds__(32)
void qlstm_recur(float* __restrict__ ws) {
    const float* Z = ws + WS_Z;
    float* H = ws + WS_H;
    const int l  = threadIdx.x;
    const float sl = ws[WS_S + l];

    float h = 0.f, c = 0.f;
    for (int t = 0; t < T_STEPS; ++t) {
        if (t + 64 < T_STEPS)
            __builtin_prefetch(Z + (size_t)(t + 64) * NG + l, 0, 3);

        float ang = Z[(size_t)t * NG + l] + h * sl;   // bias folded in Z
        float cc  = __cosf(0.5f * ang);
        float p   = cc * cc;
        p *= __shfl_xor(p, 1, 8);
        p *= __shfl_xor(p, 2, 8);
        p *= __shfl_xor(p, 4, 8);
        float q = 2.f * p - 1.f;

        float qf = __shfl(q,  0, 32);
        float qi = __shfl(q,  8, 32);
        float qu = __shfl(q, 16, 32);
        float qo = __shfl(q, 24, 32);

        float f = 1.f / (1.f + __expf(-qf));
        float i = 1.f / (1.f + __expf(-qi));
        float g = 1.f - 2.f / (__expf(2.f * qu) + 1.f);   // tanh
        float o = 1.f / (1.f + __expf(-qo));

        c = f * c + i * g;
        float th = 1.f - 2.f / (__expf(2.f * c) + 1.f);   // tanh(c)
        h = o * th;
        if (l == 0) H[t] = h;
    }
    if (l == 0) ws[WS_C] = c;
}

// ---------------------------------------------------------------------------
// Writeout: row r < T -> stacked[t]=H[t] broadcast; r==T -> hx; r==T+1 -> cx.
// 2048 floats per row = 512 float4; 256 threads x 2 stores.
// ---------------------------------------------------------------------------
__global__ __launch_bounds__(256)
void qlstm_write(const float* __restrict__ ws, float* __restrict__ out) {
    int r = blockIdx.x;                      // 0 .. T_STEPS+1
    float val;
    if (r < T_STEPS)            val = ws[WS_H + r];
    else if (r == T_STEPS)      val = ws[WS_H + T_STEPS - 1];   // hx
    else                        val = ws[WS_C];                 // cx
    float4 v4 = make_float4(val, val, val, val);
    float4* dst = (float4*)(out + (size_t)r * D_H);
    dst[threadIdx.x]       = v4;
    dst[threadIdx.x + 256] = v4;
}

// ---------------------------------------------------------------------------
extern "C" void kernel_launch(void* const* d_in, const int* in_sizes, int n_in,
                              void* d_out, int out_size, void* d_ws, size_t ws_size,
                              hipStream_t stream) {
    const float* X  = (const float*)d_in[0];
    const float* Wf = (const float*)d_in[1];
    const float* bf = (const float*)d_in[2];
    const float* Wi = (const float*)d_in[3];
    const float* bi = (const float*)d_in[4];
    const float* Wu = (const float*)d_in[5];
    const float* bu = (const float*)d_in[6];
    const float* Wo = (const float*)d_in[7];
    const float* bo = (const float*)d_in[8];
    float* ws  = (float*)d_ws;
    float* out = (float*)d_out;

    qlstm_prep<<<NG, 256, 0, stream>>>(Wf, bf, Wi, bi, Wu, bu, Wo, bo, ws);
    qlstm_gemm<<<T_STEPS / 16, 128, 0, stream>>>(X, ws);
    qlstm_recur<<<1, 32, 0, stream>>>(ws);
    qlstm_write<<<T_STEPS + 2, 256, 0, stream>>>(ws, out);
}